// InterGraphAttention_3478923509962
// MI455X (gfx1250) — compile-verified
//
#include <hip/hip_runtime.h>
#include <math.h>

// ---------------------------------------------------------------------------
// InterGraphAttention (bipartite GATConv both directions) for MI455X / gfx1250
//   N=50000 nodes/side, E=1e6 edges, D=256, H=2, C=32 (HC=64)
// Dense projections: V_WMMA_F32_16X16X4_F32 (full fp32), A staged to LDS with
// global_load_async_to_lds_b128 (ASYNCcnt path), B from pre-transposed W
// (contiguous b64 loads). Sparse softmax-aggregation: L2-resident f32 atomics
// with per-edge coefficients computed exactly once.
// ---------------------------------------------------------------------------

typedef __attribute__((ext_vector_type(2))) float v2f;
typedef __attribute__((ext_vector_type(8))) float v8f;

#define DIN 256
#define HC 64
#define LPITCH 260   // dwords; 260%64==4 -> banks 4m+k conflict-free; 260*4%16==0 -> b128-aligned rows

__device__ __forceinline__ float elu_f(float x)   { return x > 0.f ? x : (expf(x) - 1.f); }
__device__ __forceinline__ float lrelu_f(float x) { return x > 0.f ? x : 0.2f * x; }

// monotonic float<->uint mapping so atomicMax(uint) == float max (handles negatives)
__device__ __forceinline__ unsigned fenc(float f) {
  unsigned u = __float_as_uint(f);
  return (u & 0x80000000u) ? ~u : (u | 0x80000000u);
}
__device__ __forceinline__ float fdec(unsigned u) {
  return __uint_as_float((u & 0x80000000u) ? (u & 0x7FFFFFFFu) : ~u);
}

// ---- init: out=bias, amax=0 (below enc(-inf)), denom=0, and transpose both W -------
__global__ void ig_init(unsigned* __restrict__ amax, float* __restrict__ denom,
                        float* __restrict__ out, const float* __restrict__ bias,
                        const float* __restrict__ Wsrc, const float* __restrict__ Wdst,
                        float* __restrict__ WsrcT, float* __restrict__ WdstT,
                        int N) {
  int gid = blockIdx.x * blockDim.x + threadIdx.x;
  if (gid < 4 * N) {            // 2 dirs * N * 2 heads
    amax[gid]  = 0u;
    denom[gid] = 0.f;
  }
  if (gid < 2 * DIN * HC) {     // transpose [256,64] -> [64,256], both weights
    int a = gid >= DIN * HC;
    int r = gid - a * DIN * HC;
    int k = r / HC, c = r % HC;
    const float* W  = a ? Wdst  : Wsrc;
    float*       WT = a ? WdstT : WsrcT;
    WT[c * DIN + k] = W[k * HC + c];
  }
  if (gid < 2 * N * HC) out[gid] = bias[gid & (HC - 1)];
}

// ---- fused elu + dual GEMM via fp32 WMMA -----------------------------------
// Block = 256 threads = 8 waves owning 16 rows; waves 0-3 -> Wsrc cols,
// waves 4-7 -> Wdst cols; each wave builds one 16x16 tile from 64 K=4 WMMAs.
__global__ void __launch_bounds__(256)
ig_gemm_elu(const float* __restrict__ X,
            const float* __restrict__ WsrcT, const float* __restrict__ WdstT,
            float* __restrict__ xs, float* __restrict__ xd, int N) {
  __shared__ float lx[16 * LPITCH];
  const int tid  = threadIdx.x;
  const int row0 = blockIdx.x * 16;
  const bool full = (row0 + 16 <= N);   // block-uniform

  if (full) {
    // async-copy 16x256 tile: 4 iters x (256 lanes x 16B), then elu in place
    #pragma unroll
    for (int j = 0; j < 4; ++j) {
      int f  = j * 1024 + tid * 4;
      int r  = f >> 8;
      int cc = f & 255;
      const float* g = X + (long long)(row0 + r) * DIN + cc;
      unsigned lo = (unsigned)(size_t)(const void*)&lx[r * LPITCH + cc];
      asm volatile("global_load_async_to_lds_b128 %0, %1, off"
                   :: "v"(lo), "v"(g) : "memory");
    }
    asm volatile("s_wait_asynccnt 0x0" ::: "memory");
    __syncthreads();
    #pragma unroll
    for (int i = 0; i < 16; ++i) {
      float v = lx[i * LPITCH + tid];
      lx[i * LPITCH + tid] = elu_f(v);
    }
  } else {
    for (int i = 0; i < 16; ++i) {
      int r = row0 + i;
      float v = (r < N) ? X[(long long)r * DIN + tid] : 0.f;
      lx[i * LPITCH + tid] = elu_f(v);
    }
  }
  __syncthreads();

  const int lane = tid & 31;
  const int wave = tid >> 5;
  const int m    = lane & 15;   // tile row (A) / tile col (B, D)
  const int kh   = lane >> 4;   // lane-half: K-pair select (A/B), M offset (D)
  const float* WT = (wave < 4) ? WsrcT : WdstT;
  float*       O  = (wave < 4) ? xs    : xd;
  const int colg  = ((wave & 3) * 16) + m;          // output column (0..63)
  const float* Wrow = WT + (long long)colg * DIN;   // this lane's B column

  v8f acc = {};
  #pragma unroll 4
  for (int kb = 0; kb < DIN; kb += 4) {
    const int k0 = kb + 2 * kh;
    v2f a, b;
    a.x = lx[m * LPITCH + k0];        // A 16x4: lanes 0-15 K=0,1; 16-31 K=2,3
    a.y = lx[m * LPITCH + k0 + 1];
    b = *(const v2f*)&Wrow[k0];       // contiguous b64 (transposed W)
    acc = __builtin_amdgcn_wmma_f32_16x16x4_f32(
        false, a, false, b, (short)0, acc, false, false);
  }

  // D: VGPR r -> M = r + 8*kh, N = lane&15
  if (full) {
    #pragma unroll
    for (int r = 0; r < 8; ++r)
      O[(long long)(row0 + r + 8 * kh) * HC + colg] = acc[r];
  } else {
    #pragma unroll
    for (int r = 0; r < 8; ++r) {
      int mm = row0 + r + 8 * kh;
      if (mm < N) O[(long long)mm * HC + colg] = acc[r];
    }
  }
}

// ---- attention dot: a[n,h] = sum_c P[n,h,c]*att[h,c] -----------------------
__global__ void ig_att(const float* __restrict__ P, const float* __restrict__ att,
                       float* __restrict__ A, int N) {
  int gid = blockIdx.x * blockDim.x + threadIdx.x;
  if (gid >= 2 * N) return;
  int n = gid >> 1, h = gid & 1;
  const float* row = P + (long long)n * HC + h * 32;
  const float* av  = att + h * 32;
  float s = 0.f;
  #pragma unroll
  for (int c = 0; c < 32; ++c) s += row[c] * av[c];
  A[gid] = s;
}

// edge decode helper: dir 0 (t_rep): s=src,t=dst; dir 1 (h_rep): swapped;
// e in [E,E+N) are self loops.
__device__ __forceinline__ void edge_st(const int* src, const int* dst,
                                        int e, int E, int dir, int& s, int& t) {
  if (e < E) { int a = src[e], b = dst[e]; s = dir ? b : a; t = dir ? a : b; }
  else       { s = t = e - E; }
}

// ---- pass 1: segment max of leaky-relu scores ------------------------------
__global__ void ig_pass1(const int* __restrict__ src, const int* __restrict__ dst,
                         const float* __restrict__ as0, const float* __restrict__ ad0,
                         const float* __restrict__ as1, const float* __restrict__ ad1,
                         unsigned* __restrict__ amax, int E, int N) {
  int gid = blockIdx.x * blockDim.x + threadIdx.x;
  int M = E + N;
  if (gid >= 2 * M * 2) return;
  int h = gid & 1, rest = gid >> 1;
  int dir = rest >= M, e = rest - dir * M;
  int s, t; edge_st(src, dst, e, E, dir, s, t);
  const float* AS = dir ? as1 : as0;
  const float* AD = dir ? ad1 : ad0;
  float alpha = lrelu_f(AS[s * 2 + h] + AD[t * 2 + h]);
  atomicMax(&amax[(dir * N + t) * 2 + h], fenc(alpha));
}

// ---- pass 2: denom = segment sum of exp(a - amax) --------------------------
__global__ void ig_pass2(const int* __restrict__ src, const int* __restrict__ dst,
                         const float* __restrict__ as0, const float* __restrict__ ad0,
                         const float* __restrict__ as1, const float* __restrict__ ad1,
                         const unsigned* __restrict__ amax, float* __restrict__ denom,
                         int E, int N) {
  int gid = blockIdx.x * blockDim.x + threadIdx.x;
  int M = E + N;
  if (gid >= 2 * M * 2) return;
  int h = gid & 1, rest = gid >> 1;
  int dir = rest >= M, e = rest - dir * M;
  int s, t; edge_st(src, dst, e, E, dir, s, t);
  const float* AS = dir ? as1 : as0;
  const float* AD = dir ? ad1 : ad0;
  int idx = (dir * N + t) * 2 + h;
  float alpha = lrelu_f(AS[s * 2 + h] + AD[t * 2 + h]);
  atomicAdd(&denom[idx], expf(alpha - fdec(amax[idx])));
}

// ---- pass 2b: final per-edge softmax coefficient (computed once) -----------
__global__ void ig_coef(const int* __restrict__ src, const int* __restrict__ dst,
                        const float* __restrict__ as0, const float* __restrict__ ad0,
                        const float* __restrict__ as1, const float* __restrict__ ad1,
                        const unsigned* __restrict__ amax, const float* __restrict__ denom,
                        float* __restrict__ coef, int E, int N) {
  int gid = blockIdx.x * blockDim.x + threadIdx.x;
  int M = E + N;
  if (gid >= 2 * M * 2) return;
  int h = gid & 1, rest = gid >> 1;
  int dir = rest >= M, e = rest - dir * M;
  int s, t; edge_st(src, dst, e, E, dir, s, t);
  const float* AS = dir ? as1 : as0;
  const float* AD = dir ? ad1 : ad0;
  int idx = (dir * N + t) * 2 + h;
  float alpha = lrelu_f(AS[s * 2 + h] + AD[t * 2 + h]);
  coef[gid] = expf(alpha - fdec(amax[idx])) / (denom[idx] + 1e-16f);
}

// ---- pass 3: normalized message scatter ------------------------------------
// thread per (dir, edge, channel): out[t,c] += coef[dir,e,h] * xs[s,c]
__global__ void ig_pass3(const int* __restrict__ src, const int* __restrict__ dst,
                         const float* __restrict__ xs0, const float* __restrict__ xs1,
                         const float* __restrict__ coef,
                         float* __restrict__ outT, float* __restrict__ outH,
                         int E, int N) {
  long long gid = (long long)blockIdx.x * blockDim.x + threadIdx.x;
  int M = E + N;
  if (gid >= (long long)2 * M * HC) return;
  int c    = (int)(gid & (HC - 1));
  int rest = (int)(gid >> 6);
  int dir  = rest >= M, e = rest - dir * M;
  int h    = c >> 5;
  int s, t; edge_st(src, dst, e, E, dir, s, t);
  const float* XS = dir ? xs1 : xs0;
  float*      OUT = dir ? outH : outT;
  float cf = coef[rest * 2 + h];     // rest == dir*M + e
  atomicAdd(&OUT[(long long)t * HC + c], cf * XS[(long long)s * HC + c]);
}

// ---------------------------------------------------------------------------
extern "C" void kernel_launch(void* const* d_in, const int* in_sizes, int n_in,
                              void* d_out, int out_size, void* d_ws, size_t ws_size,
                              hipStream_t stream) {
  const float* h_x     = (const float*)d_in[0];
  const float* t_x     = (const float*)d_in[1];
  const int*   eidx    = (const int*)d_in[2];
  const float* Wsrc    = (const float*)d_in[3];
  const float* Wdst    = (const float*)d_in[4];
  const float* att_src = (const float*)d_in[5];
  const float* att_dst = (const float*)d_in[6];
  const float* bias    = (const float*)d_in[7];

  const int N = in_sizes[0] / DIN;
  const int E = in_sizes[2] / 2;
  const int M = E + N;
  const int* src = eidx;
  const int* dst = eidx + E;

  float* outH = (float*)d_out;                 // h_rep first in return order
  float* outT = outH + (long long)N * HC;      // then t_rep

  // workspace carve (~72 MB)
  float* ws    = (float*)d_ws;
  float* xs_h  = ws; ws += (long long)N * HC;  // elu(h) @ Wsrc
  float* xd_h  = ws; ws += (long long)N * HC;  // elu(h) @ Wdst
  float* xs_t  = ws; ws += (long long)N * HC;  // elu(t) @ Wsrc
  float* xd_t  = ws; ws += (long long)N * HC;  // elu(t) @ Wdst
  float* a_sh  = ws; ws += 2 * N;
  float* a_dh  = ws; ws += 2 * N;
  float* a_st  = ws; ws += 2 * N;
  float* a_dt  = ws; ws += 2 * N;
  unsigned* amax = (unsigned*)ws; ws += 4 * N; // [2 dirs][N][2 heads]
  float* denom = ws; ws += 4 * N;
  float* WsrcT = ws; ws += DIN * HC;           // [64,256]
  float* WdstT = ws; ws += DIN * HC;
  float* coef  = ws; ws += (long long)2 * M * 2;

  const int initN = 2 * N * HC;
  ig_init<<<(initN + 255) / 256, 256, 0, stream>>>(amax, denom, outH, bias,
                                                   Wsrc, Wdst, WsrcT, WdstT, N);

  const int gb = (N + 15) / 16;
  ig_gemm_elu<<<gb, 256, 0, stream>>>(h_x, WsrcT, WdstT, xs_h, xd_h, N);
  ig_gemm_elu<<<gb, 256, 0, stream>>>(t_x, WsrcT, WdstT, xs_t, xd_t, N);

  const int ab = (2 * N + 127) / 128;
  ig_att<<<ab, 128, 0, stream>>>(xs_h, att_src, a_sh, N);
  ig_att<<<ab, 128, 0, stream>>>(xd_h, att_dst, a_dh, N);
  ig_att<<<ab, 128, 0, stream>>>(xs_t, att_src, a_st, N);
  ig_att<<<ab, 128, 0, stream>>>(xd_t, att_dst, a_dt, N);

  const int p12 = 2 * M * 2;
  ig_pass1<<<(p12 + 255) / 256, 256, 0, stream>>>(src, dst, a_sh, a_dt, a_st, a_dh,
                                                  amax, E, N);
  ig_pass2<<<(p12 + 255) / 256, 256, 0, stream>>>(src, dst, a_sh, a_dt, a_st, a_dh,
                                                  amax, denom, E, N);
  ig_coef<<<(p12 + 255) / 256, 256, 0, stream>>>(src, dst, a_sh, a_dt, a_st, a_dh,
                                                 amax, denom, coef, E, N);
  const long long p3 = (long long)2 * M * HC;
  ig_pass3<<<(unsigned)((p3 + 255) / 256), 256, 0, stream>>>(
      src, dst, xs_h, xs_t, coef, outT, outH, E, N);
}